// DHDeePC_481036337158
// MI455X (gfx1250) — compile-verified
//
#include <hip/hip_runtime.h>
#include <cstddef>
#include <cstdint>

// ---------------- problem constants ----------------
#define CM    4
#define CP    6
#define CTINI 50
#define CNH   100
#define CTT   2000
#define CNB   4096
#define CL    1851           // TT - TINI - NH + 1
#define CLP   1856           // L padded to multiple of 64
#define CNC   506            // TINI*M + TINI*P + P
#define CNCP  512            // NC padded
#define KY    608            // Yf K-dim padded (600 -> mult of 32)
#define KUP   416            // Uf K-dim padded (400 -> mult of 32)
#define YFROWS 640           // Yf rows padded to mult of 64 (M-dim use)
#define UFROWS 448           // Uf rows padded to mult of 64
#define KCH   32             // GEMM K-chunk staged per LDS buffer

typedef __attribute__((ext_vector_type(2))) float v2f;
typedef __attribute__((ext_vector_type(8))) float v8f;

// -------- CDNA5 async global->LDS staging (ASYNCcnt path), with fallback ----
#if defined(__has_builtin)
#if __has_builtin(__builtin_amdgcn_global_load_async_to_lds_b32)
#define USE_ASYNC_LDS 1
#endif
#endif

#ifdef USE_ASYNC_LDS
__device__ __forceinline__ void async_g2l(const float* g, float* l)
{
    // (global src, lds dst, imm offset, imm cpol)
    __builtin_amdgcn_global_load_async_to_lds_b32(
        (__attribute__((address_space(1))) int*)(uintptr_t)g,
        (__attribute__((address_space(3))) int*)(uint32_t)(uintptr_t)l,
        0, 0);
}
#if __has_builtin(__builtin_amdgcn_s_wait_asynccnt)
#define WAIT_ASYNC(n) __builtin_amdgcn_s_wait_asynccnt(n)
#else
#define WAIT_ASYNC(n) asm volatile("s_wait_asynccnt %0" ::"i"(n))
#endif
#endif

// =====================================================================
// Tiled fp32 WMMA GEMM:  C[M x N] = alpha * A[M x K] * B[K x N] + beta*C
// Row-major, arbitrary leading dims. M,N multiples of 64; K multiple of 32.
// Workgroup = 128 threads (4 waves), 64x64 C tile, wave = 32x32 (2x2 frags).
// Double-buffered LDS staged by async global->LDS copies (s_wait_asynccnt),
// inner product via v_wmma_f32_16x16x4_f32 (32 WMMAs per 32-K chunk).
// =====================================================================
__global__ __launch_bounds__(128) void gemm64_k(
    const float* __restrict__ A, int lda,
    const float* __restrict__ B, int ldb,
    float* __restrict__ C, int ldc,
    int K, float alpha, float beta)
{
    __shared__ float As[2][64][33];   // 64 rows x 32 k (pad 33: no bank conflicts)
    __shared__ float Bs[2][32][64];   // 32 k x 64 cols

    const int t    = threadIdx.x;
    const int lane = t & 31;
    const int wave = t >> 5;              // 0..3
    const int wm   = (wave >> 1) * 32;    // wave row offset in tile
    const int wn   = (wave & 1) * 32;     // wave col offset in tile
    const int m16  = lane & 15;
    const int hi   = lane >> 4;           // 0/1 half-wave

    const size_t tm0 = (size_t)blockIdx.y * 64;
    const size_t tn0 = (size_t)blockIdx.x * 64;
    const int nch = K / KCH;

    v8f acc[2][2] = {};

    auto stage = [&](int ch, int buf) {
        const size_t kb = (size_t)ch * KCH;
#pragma unroll
        for (int i = 0; i < 16; ++i) {           // A chunk: 64 x 32
            int idx = t + i * 128;
            int r = idx >> 5, c = idx & 31;
            const float* g = &A[(tm0 + (size_t)r) * (size_t)lda + kb + (size_t)c];
#ifdef USE_ASYNC_LDS
            async_g2l(g, &As[buf][r][c]);
#else
            As[buf][r][c] = *g;
#endif
        }
#pragma unroll
        for (int i = 0; i < 16; ++i) {           // B chunk: 32 x 64
            int idx = t + i * 128;
            int r = idx >> 6, c = idx & 63;
            const float* g = &B[(kb + (size_t)r) * (size_t)ldb + tn0 + (size_t)c];
#ifdef USE_ASYNC_LDS
            async_g2l(g, &Bs[buf][r][c]);
#else
            Bs[buf][r][c] = *g;
#endif
        }
    };

    stage(0, 0);
    for (int ch = 0; ch < nch; ++ch) {
        const int buf = ch & 1;
        const bool has_next = (ch + 1) < nch;
        if (has_next) stage(ch + 1, buf ^ 1);
#ifdef USE_ASYNC_LDS
        if (has_next) WAIT_ASYNC(32);   // chunk `ch` (32 async ops older) done
        else          WAIT_ASYNC(0);
#endif
        __syncthreads();

#pragma unroll
        for (int kk = 0; kk < 8; ++kk) {
            const int k0 = kk * 4;
            v2f a[2], b[2];
#pragma unroll
            for (int i = 0; i < 2; ++i) {
                int row = wm + i * 16 + m16;
                a[i].x = As[buf][row][k0 + 2 * hi];
                a[i].y = As[buf][row][k0 + 2 * hi + 1];
            }
#pragma unroll
            for (int j = 0; j < 2; ++j) {
                int col = wn + j * 16 + m16;
                b[j].x = Bs[buf][k0 + 2 * hi][col];
                b[j].y = Bs[buf][k0 + 2 * hi + 1][col];
            }
#pragma unroll
            for (int i = 0; i < 2; ++i)
#pragma unroll
                for (int j = 0; j < 2; ++j)
                    acc[i][j] = __builtin_amdgcn_wmma_f32_16x16x4_f32(
                        false, a[i], false, b[j], (short)0, acc[i][j], false, false);
        }
        __syncthreads();
    }

    // epilogue: lane holds C rows (r + 8*hi), col m16 for each 16x16 fragment
#pragma unroll
    for (int i = 0; i < 2; ++i)
#pragma unroll
        for (int j = 0; j < 2; ++j)
#pragma unroll
            for (int r = 0; r < 8; ++r) {
                size_t row = tm0 + (size_t)(wm + i * 16 + r + 8 * hi);
                size_t col = tn0 + (size_t)(wn + j * 16 + m16);
                float v = alpha * acc[i][j][r];
                if (beta != 0.0f) v += beta * C[row * (size_t)ldc + col];
                C[row * (size_t)ldc + col] = v;
            }
}

static inline void gemm_launch(hipStream_t s,
                               const float* A, int lda,
                               const float* B, int ldb,
                               float* C, int ldc,
                               int M, int N, int K, float alpha, float beta)
{
    dim3 g(N / 64, M / 64);
    gemm64_k<<<g, 128, 0, s>>>(A, lda, B, ldb, C, ldc, K, alpha, beta);
}

// =====================================================================
// 64x64 in-place Cholesky of a diagonal block (lower), upper zeroed.
// =====================================================================
__global__ __launch_bounds__(64) void chol_diag_k(float* D, int ld)
{
    __shared__ float a[64][65];
    const int t = threadIdx.x;
    for (int c = 0; c < 64; ++c) a[t][c] = D[(size_t)t * ld + c];
    __syncthreads();
    for (int j = 0; j < 64; ++j) {
        if (t == j) a[j][j] = sqrtf(a[j][j]);
        __syncthreads();
        if (t > j) a[t][j] /= a[j][j];
        __syncthreads();
        if (t > j)
            for (int c = j + 1; c <= t; ++c) a[t][c] -= a[t][j] * a[c][j];
        __syncthreads();
    }
    for (int c = 0; c < 64; ++c) D[(size_t)t * ld + c] = (c <= t) ? a[t][c] : 0.0f;
}

// Panel TRSM of Cholesky: P (nrows x 64) <- P * Lkk^-T  (one thread per row)
__global__ __launch_bounds__(64) void chol_panel_k(const float* __restrict__ Lkk, int ld,
                                                   float* __restrict__ P, int nrows)
{
    __shared__ float Lk[64][65];
    __shared__ float xs[64][64];
    const int t = threadIdx.x;
    for (int c = 0; c < 64; ++c) Lk[t][c] = Lkk[(size_t)t * ld + c];
    __syncthreads();
    const int row = blockIdx.x * 64 + t;
    if (row < nrows) {
        float* ap = P + (size_t)row * ld;
        for (int j = 0; j < 64; ++j) {
            float ssum = ap[j];
            for (int c = 0; c < j; ++c) ssum -= xs[c][t] * Lk[j][c];
            xs[j][t] = ssum / Lk[j][j];
        }
        for (int j = 0; j < 64; ++j) ap[j] = xs[j][t];
    }
}

// Solve Lkk * Z = B (64 x ncols), in place. One thread per column.
__global__ __launch_bounds__(64) void trsm_lower_k(const float* __restrict__ Lkk, int ld,
                                                   float* __restrict__ B, int ldb)
{
    __shared__ float Lk[64][65];
    __shared__ float xs[64][64];
    const int t = threadIdx.x;
    for (int c = 0; c < 64; ++c) Lk[t][c] = Lkk[(size_t)t * ld + c];
    __syncthreads();
    const size_t col = (size_t)blockIdx.x * 64 + t;
    for (int j = 0; j < 64; ++j) {
        float ssum = B[(size_t)j * ldb + col];
        for (int c = 0; c < j; ++c) ssum -= Lk[j][c] * xs[c][t];
        xs[j][t] = ssum / Lk[j][j];
    }
    for (int j = 0; j < 64; ++j) B[(size_t)j * ldb + col] = xs[j][t];
}

// Solve Ukk * Z = B (upper triangular 64x64), in place.
__global__ __launch_bounds__(64) void trsm_upper_k(const float* __restrict__ Ukk, int ld,
                                                   float* __restrict__ B, int ldb)
{
    __shared__ float Uk[64][65];
    __shared__ float xs[64][64];
    const int t = threadIdx.x;
    for (int c = 0; c < 64; ++c) Uk[t][c] = Ukk[(size_t)t * ld + c];
    __syncthreads();
    const size_t col = (size_t)blockIdx.x * 64 + t;
    for (int j = 63; j >= 0; --j) {
        float ssum = B[(size_t)j * ldb + col];
        for (int c = j + 1; c < 64; ++c) ssum -= Uk[j][c] * xs[c][t];
        xs[j][t] = ssum / Uk[j][j];
    }
    for (int j = 0; j < 64; ++j) B[(size_t)j * ldb + col] = xs[j][t];
}

// generic tiled transpose: out[c][r] = in[r][c]
__global__ void transpose_k(const float* __restrict__ in, int ldin,
                            float* __restrict__ out, int ldout,
                            int rows, int cols)
{
    __shared__ float tile[32][33];
    const int c0 = blockIdx.x * 32, r0 = blockIdx.y * 32;
    const int tx = threadIdx.x, ty = threadIdx.y;  // (32,8)
    for (int i = 0; i < 32; i += 8) {
        int r = r0 + ty + i, c = c0 + tx;
        if (r < rows && c < cols) tile[ty + i][tx] = in[(size_t)r * ldin + c];
    }
    __syncthreads();
    for (int i = 0; i < 32; i += 8) {
        int r = c0 + ty + i, c = r0 + tx;  // output coords
        if (r < cols && c < rows) out[(size_t)r * ldout + c] = tile[tx][ty + i];
    }
}

// ---------------- Hankel-slice / RHS build kernels ----------------
__global__ void build_Uf_k(const float* __restrict__ ud, float* __restrict__ Uf)
{
    int idx = blockIdx.x * blockDim.x + threadIdx.x;
    const int n = UFROWS * CLP;
    if (idx >= n) return;
    int r = idx / CLP, j = idx % CLP;
    float v = 0.0f;
    if (r < CNH * CM && j < CL) v = ud[(size_t)(CTINI + r / CM + j) * CM + (r % CM)];
    Uf[idx] = v;
}
__global__ void build_Yf_k(const float* __restrict__ yd, float* __restrict__ Yf)
{
    int idx = blockIdx.x * blockDim.x + threadIdx.x;
    const int n = YFROWS * CLP;
    if (idx >= n) return;
    int r = idx / CLP, j = idx % CLP;
    float v = 0.0f;
    if (r < CNH * CP && j < CL) v = yd[(size_t)(CTINI + r / CP + j) * CP + (r % CP)];
    Yf[idx] = v;
}
__global__ void build_qYf_k(const float* __restrict__ yd, const float* __restrict__ q,
                            float* __restrict__ o)
{
    int idx = blockIdx.x * blockDim.x + threadIdx.x;
    const int n = KY * CLP;
    if (idx >= n) return;
    int r = idx / CLP, j = idx % CLP;
    float v = 0.0f;
    if (r < CNH * CP && j < CL) v = q[r] * yd[(size_t)(CTINI + r / CP + j) * CP + (r % CP)];
    o[idx] = v;
}
__global__ void build_rUf_k(const float* __restrict__ ud, const float* __restrict__ rv,
                            float* __restrict__ o)
{
    int idx = blockIdx.x * blockDim.x + threadIdx.x;
    const int n = KUP * CLP;
    if (idx >= n) return;
    int r = idx / CLP, j = idx % CLP;
    float v = 0.0f;
    if (r < CNH * CM && j < CL) v = rv[r] * ud[(size_t)(CTINI + r / CM + j) * CM + (r % CM)];
    o[idx] = v;
}
__global__ void build_Amat_k(const float* __restrict__ ud, const float* __restrict__ yd,
                             float* __restrict__ Amat)
{
    int idx = blockIdx.x * blockDim.x + threadIdx.x;
    const int n = CNCP * CLP;
    if (idx >= n) return;
    int r = idx / CLP, j = idx % CLP;
    float v = 0.0f;
    if (j < CL) {
        if (r < CTINI * CM) {                       // Up
            v = ud[(size_t)(r / CM + j) * CM + (r % CM)];
        } else if (r < CTINI * CM + CTINI * CP) {   // Yp
            int rr = r - CTINI * CM;
            v = yd[(size_t)(rr / CP + j) * CP + (rr % CP)];
        } else if (r < CNC) {                       // Yf last P rows
            int k = r - (CTINI * CM + CTINI * CP);
            v = yd[(size_t)(CTINI + CNH - 1 + j) * CP + k];
        }
    }
    Amat[idx] = v;
}
__global__ void build_qrefT_k(const float* __restrict__ ref, const float* __restrict__ q,
                              float* __restrict__ o)
{
    int idx = blockIdx.x * blockDim.x + threadIdx.x;
    const int n = KY * CNB;
    if (idx >= n) return;
    int r = idx / CNB, b = idx % CNB;
    float v = 0.0f;
    if (r < CNH * CP) v = q[r] * ref[(size_t)b * (CNH * CP) + r];
    o[idx] = v;
}
__global__ void build_b2_k(const float* __restrict__ u_ini, const float* __restrict__ y_ini,
                           const float* __restrict__ ref, float* __restrict__ R)
{
    int idx = blockIdx.x * blockDim.x + threadIdx.x;
    const int n = CNCP * CNB;
    if (idx >= n) return;
    int r = idx / CNB, b = idx % CNB;
    float v = 0.0f;
    if (r < CTINI * CM) v = u_ini[(size_t)b * (CTINI * CM) + r];
    else if (r < CTINI * CM + CTINI * CP) v = y_ini[(size_t)b * (CTINI * CP) + (r - CTINI * CM)];
    else if (r < CNC) v = ref[(size_t)b * (CNH * CP) + (CNH * CP - CP) + (r - (CTINI * CM + CTINI * CP))];
    R[idx] = v;
}
// add 'addv' to diag[0..nreal), set 1.0 on pad diag [nreal..ntot)
__global__ void add_diag_k(float* __restrict__ Mp, int ld, int nreal, int ntot, float addv)
{
    int i = blockIdx.x * blockDim.x + threadIdx.x;
    if (i >= ntot) return;
    if (i < nreal) Mp[(size_t)i * ld + i] += addv;
    else           Mp[(size_t)i * ld + i] = 1.0f;
}

// ---------------- host-side composite routines (all on stream) ----------------
static inline void transpose_launch(hipStream_t s, const float* in, int ldin,
                                    float* out, int ldout, int rows, int cols)
{
    dim3 g((cols + 31) / 32, (rows + 31) / 32);
    transpose_k<<<g, dim3(32, 8), 0, s>>>(in, ldin, out, ldout, rows, cols);
}

// Blocked Cholesky in place (lower), n multiple of 64. panT: >= 64*n floats.
static inline void chol_inplace(hipStream_t s, float* G, int n, int ld, float* panT)
{
    const int nb = n / 64;
    for (int k = 0; k < nb; ++k) {
        float* Dk = G + (size_t)(k * 64) * ld + (size_t)(k * 64);
        chol_diag_k<<<1, 64, 0, s>>>(Dk, ld);
        const int nrem = n - (k + 1) * 64;
        if (nrem > 0) {
            float* P = G + (size_t)((k + 1) * 64) * ld + (size_t)(k * 64);
            chol_panel_k<<<nrem / 64, 64, 0, s>>>(Dk, ld, P, nrem);
            // trailing SYRK: A22 -= L21 * L21^T  (via explicit transpose + WMMA GEMM)
            transpose_launch(s, P, ld, panT, nrem, nrem, 64);
            float* C22 = G + (size_t)((k + 1) * 64) * ld + (size_t)((k + 1) * 64);
            gemm_launch(s, P, ld, panT, nrem, C22, ld, nrem, nrem, 64, -1.0f, 1.0f);
        }
    }
}

// Forward solve  L * Z = B  (n x ncols), in place. L = lower triangle of Lfull.
static inline void trsm_lower_all(hipStream_t s, const float* Lfull, int ld, int n,
                                  float* B, int ldb, int ncols)
{
    const int nb = n / 64;
    for (int k = 0; k < nb; ++k) {
        if (k > 0)
            gemm_launch(s, Lfull + (size_t)(k * 64) * ld, ld, B, ldb,
                        B + (size_t)(k * 64) * ldb, ldb, 64, ncols, k * 64, -1.0f, 1.0f);
        trsm_lower_k<<<ncols / 64, 64, 0, s>>>(
            Lfull + (size_t)(k * 64) * ld + (size_t)(k * 64), ld,
            B + (size_t)(k * 64) * ldb, ldb);
    }
}

// Backward solve  U * Z = B  with U = Lfull^T stored explicitly (upper).
static inline void trsm_upper_all(hipStream_t s, const float* Ufull, int ld, int n,
                                  float* B, int ldb, int ncols)
{
    const int nb = n / 64;
    for (int k = nb - 1; k >= 0; --k) {
        const int tail = n - (k + 1) * 64;
        if (tail > 0)
            gemm_launch(s, Ufull + (size_t)(k * 64) * ld + (size_t)((k + 1) * 64), ld,
                        B + (size_t)((k + 1) * 64) * ldb, ldb,
                        B + (size_t)(k * 64) * ldb, ldb, 64, ncols, tail, -1.0f, 1.0f);
        trsm_upper_k<<<ncols / 64, 64, 0, s>>>(
            Ufull + (size_t)(k * 64) * ld + (size_t)(k * 64), ld,
            B + (size_t)(k * 64) * ldb, ldb);
    }
}

#define LAUNCH1D(kern, n, ...) kern<<<((n) + 255) / 256, 256, 0, stream>>>(__VA_ARGS__)

// =====================================================================
extern "C" void kernel_launch(void* const* d_in, const int* in_sizes, int n_in,
                              void* d_out, int out_size, void* d_ws, size_t ws_size,
                              hipStream_t stream)
{
    (void)in_sizes; (void)n_in; (void)out_size; (void)ws_size;
    const float* ud    = (const float*)d_in[0];  // [2000,4]
    const float* yd    = (const float*)d_in[1];  // [2000,6]
    const float* q     = (const float*)d_in[2];  // [600]
    const float* rv    = (const float*)d_in[3];  // [400]
    const float* ref   = (const float*)d_in[4];  // [4096,600]
    const float* u_ini = (const float*)d_in[5];  // [4096,200]
    const float* y_ini = (const float*)d_in[6];  // [4096,300]
    float* out = (float*)d_out;                  // inp(400x4096) ++ out(600x4096)

    // -------- workspace carving (floats); total ~128 MB --------
    float* ws = (float*)d_ws;
    size_t off = 0;
    auto carve = [&](size_t nelem) { float* p = ws + off; off += nelem; return p; };
    float* Uf    = carve((size_t)UFROWS * CLP);   // 448 x 1856
    float* Yf    = carve((size_t)YFROWS * CLP);   // 640 x 1856
    float* qYf   = carve((size_t)KY * CLP);       // 608 x 1856
    float* rUf   = carve((size_t)KUP * CLP);      // 416 x 1856
    float* UfT   = carve((size_t)CLP * UFROWS);   // 1856 x 448  (ld 448)
    float* YfT   = carve((size_t)CLP * YFROWS);   // 1856 x 640  (ld 640)
    float* G     = carve((size_t)CLP * CLP);      // 2H, then Cholesky L
    float* LT    = carve((size_t)CLP * CLP);      // L^T
    float* B1W   = carve((size_t)CLP * CNB);      // b1 -> W -> g
    float* qrefT = carve((size_t)KY * CNB);       // 608 x 4096
    float* Amat  = carve((size_t)CNCP * CLP);     // 512 x 1856
    float* X     = carve((size_t)CLP * CNCP);     // A^T -> G^-1 A^T
    float* S     = carve((size_t)CNCP * CNCP);    // Schur complement
    float* SLT   = carve((size_t)CNCP * CNCP);
    float* R     = carve((size_t)CNCP * CNB);     // b2 -> A W - b2 -> lambda
    float* inpP  = carve((size_t)UFROWS * CNB);
    float* outP  = carve((size_t)YFROWS * CNB);
    float* panT  = carve((size_t)64 * CLP);       // Cholesky panel transpose scratch

    // -------- 1. build Hankel slices, scaled copies, RHS pieces --------
    LAUNCH1D(build_Uf_k,    UFROWS * CLP, ud, Uf);
    LAUNCH1D(build_Yf_k,    YFROWS * CLP, yd, Yf);
    LAUNCH1D(build_qYf_k,   KY * CLP,     yd, q, qYf);
    LAUNCH1D(build_rUf_k,   KUP * CLP,    ud, rv, rUf);
    LAUNCH1D(build_Amat_k,  CNCP * CLP,   ud, yd, Amat);
    LAUNCH1D(build_qrefT_k, KY * CNB,     ref, q, qrefT);
    LAUNCH1D(build_b2_k,    CNCP * CNB,   u_ini, y_ini, ref, R);

    // -------- 2. transposes --------
    transpose_launch(stream, Uf, CLP, UfT, UFROWS, UFROWS, CLP);   // 1856 x 448
    transpose_launch(stream, Yf, CLP, YfT, YFROWS, YFROWS, CLP);   // 1856 x 640
    transpose_launch(stream, Amat, CLP, X, CNCP, CNCP, CLP);       // X = A^T

    // -------- 3. G = 2 Yf^T Q Yf + 2 Uf^T R Uf + 2*delta I (pad diag = 1) ----
    gemm_launch(stream, YfT, YFROWS, qYf, CLP, G, CLP, CLP, CLP, KY, 2.0f, 0.0f);
    gemm_launch(stream, UfT, UFROWS, rUf, CLP, G, CLP, CLP, CLP, KUP, 2.0f, 1.0f);
    LAUNCH1D(add_diag_k, CLP, G, CLP, CL, CLP, 2.0e-6f);

    // -------- 4. b1 = 2 Yf^T (q o ref^T) --------
    gemm_launch(stream, YfT, YFROWS, qrefT, CNB, B1W, CNB, CLP, CNB, KY, 2.0f, 0.0f);

    // -------- 5. Cholesky of G, build L^T --------
    chol_inplace(stream, G, CLP, CLP, panT);
    transpose_launch(stream, G, CLP, LT, CLP, CLP, CLP);

    // -------- 6. X = G^-1 A^T --------
    trsm_lower_all(stream, G, CLP, CLP, X, CNCP, CNCP);
    trsm_upper_all(stream, LT, CLP, CLP, X, CNCP, CNCP);

    // -------- 7. S = A X, Cholesky(S) --------
    gemm_launch(stream, Amat, CLP, X, CNCP, S, CNCP, CNCP, CNCP, CLP, 1.0f, 0.0f);
    LAUNCH1D(add_diag_k, CNCP, S, CNCP, CNC, CNCP, 0.0f);
    chol_inplace(stream, S, CNCP, CNCP, panT);
    transpose_launch(stream, S, CNCP, SLT, CNCP, CNCP, CNCP);

    // -------- 8. W = G^-1 b1 (in place over B1W) --------
    trsm_lower_all(stream, G, CLP, CLP, B1W, CNB, CNB);
    trsm_upper_all(stream, LT, CLP, CLP, B1W, CNB, CNB);

    // -------- 9. R = A W - b2 (R pre-loaded with b2) --------
    gemm_launch(stream, Amat, CLP, B1W, CNB, R, CNB, CNCP, CNB, CLP, 1.0f, -1.0f);

    // -------- 10. lambda = S^-1 R (in place) --------
    trsm_lower_all(stream, S, CNCP, CNCP, R, CNB, CNB);
    trsm_upper_all(stream, SLT, CNCP, CNCP, R, CNB, CNB);

    // -------- 11. g = W - X lambda (in place over B1W) --------
    gemm_launch(stream, X, CNCP, R, CNB, B1W, CNB, CLP, CNB, CNCP, -1.0f, 1.0f);

    // -------- 12. inp = Uf g ; out = Yf g --------
    gemm_launch(stream, Uf, CLP, B1W, CNB, inpP, CNB, UFROWS, CNB, CLP, 1.0f, 0.0f);
    gemm_launch(stream, Yf, CLP, B1W, CNB, outP, CNB, YFROWS, CNB, CLP, 1.0f, 0.0f);

    // -------- 13. pack outputs (padded rows are contiguous prefixes) --------
    (void)hipMemcpyAsync(out, inpP, (size_t)(CNH * CM) * CNB * sizeof(float),
                         hipMemcpyDeviceToDevice, stream);
    (void)hipMemcpyAsync(out + (size_t)(CNH * CM) * CNB, outP,
                         (size_t)(CNH * CP) * CNB * sizeof(float),
                         hipMemcpyDeviceToDevice, stream);
}